// NadarayaWatsonRegression_16312285790466
// MI455X (gfx1250) — compile-verified
//
#include <hip/hip_runtime.h>
#include <hip/hip_bf16.h>
#include <cstdint>

// ---------------------------------------------------------------------------
// Nadaraya-Watson regression, fused single-pass online softmax.
//   scores = -(q - k)^2 / (2 sigma^2);  out = softmax(scores) . value
// Memory bound: 512 MB streamed once -> ~22 us at 23.3 TB/s.
// CDNA5 path: GLOBAL_LOAD_ASYNC_TO_LDS_B128 double-buffered pipeline +
// S_WAIT_ASYNCCNT, native v_exp_f32 (exp2 domain), wave32 shuffle reduction.
// ---------------------------------------------------------------------------

#define TPB   256           // 8 wave32s
#define TILE  1024          // floats per tile per array (1 b128 / lane)

// 16-byte int vector with explicit address spaces: exact builtin param types.
typedef int i32x4 __attribute__((vector_size(16)));
typedef i32x4 __attribute__((address_space(1))) *g_v4i_ptr;   // global
typedef i32x4 __attribute__((address_space(3))) *l_v4i_ptr;   // LDS

__device__ __forceinline__ float fast_exp2(float x) {
#if __has_builtin(__builtin_amdgcn_exp2f)
  return __builtin_amdgcn_exp2f(x);          // v_exp_f32 (native exp2)
#else
  return exp2f(x);
#endif
}

// Async copy of 16 bytes/lane from global to LDS (per-lane LDS address).
__device__ __forceinline__ void async_cp16(const float4* g, float4* l) {
#if __has_builtin(__builtin_amdgcn_global_load_async_to_lds_b128)
  __builtin_amdgcn_global_load_async_to_lds_b128(
      (g_v4i_ptr)(g), (l_v4i_ptr)(l), /*offset=*/0, /*cpol=*/0);
#else
  // Fallback: raw CDNA5 encoding. vdst = LDS byte address, vaddr = 64b global.
  asm volatile("global_load_async_to_lds_b128 %0, %1, off"
               :: "v"((unsigned)(uintptr_t)l), "v"(g)
               : "memory");
#endif
}

template <int CNT>
__device__ __forceinline__ void wait_async() {
#if __has_builtin(__builtin_amdgcn_s_wait_asynccnt)
  __builtin_amdgcn_s_wait_asynccnt(CNT);
#else
  asm volatile("s_wait_asynccnt %0" :: "i"(CNT) : "memory");
#endif
  asm volatile("" ::: "memory");             // keep LDS reads behind the wait
}

// Merge two online-softmax partials (max m, weight-sum s, weighted-value v).
__device__ __forceinline__ void nw_merge(float& m, float& s, float& v,
                                         float mo, float so, float vo) {
  float nm = fmaxf(m, mo);
  float r1 = fast_exp2(m  - nm);
  float r2 = fast_exp2(mo - nm);
  s = fmaf(s, r1, so * r2);
  v = fmaf(v, r1, vo * r2);
  m = nm;
}

__global__ __launch_bounds__(TPB)
void nw_regression_kernel(const float* __restrict__ key,
                          const float* __restrict__ value,
                          const float* __restrict__ query,
                          const float* __restrict__ sigma,
                          float* __restrict__ out,
                          int N) {
  __shared__ float4 kb[2][TPB];
  __shared__ float4 vb[2][TPB];
  __shared__ float redm[TPB / 32], reds[TPB / 32], redv[TPB / 32];

  const int row = blockIdx.x;
  const int t   = threadIdx.x;

  const float q  = query[row];
  const float sg = sigma[0];
  // softmax in exp2 domain: arg = -(log2e / (2 s^2)) * d^2
  const float coef = -0.7213475204444817f / (sg * sg);

  const float4* kg = (const float4*)(key   + (size_t)row * (size_t)N);
  const float4* vg = (const float4*)(value + (size_t)row * (size_t)N);
  const int nt = N / TILE;                   // N % TILE == 0 (16384/1024 = 16)

  // Prologue: tile 0 in flight.
  async_cp16(kg + t, &kb[0][t]);
  async_cp16(vg + t, &vb[0][t]);

  float m = -__builtin_inff(), s = 0.f, v = 0.f;

  for (int j = 0; j < nt; ++j) {
    const int st = j & 1;
    if (j + 1 < nt) {
      const int sn = st ^ 1;
      const int fo = (j + 1) * (TILE / 4) + t;   // float4 offset
      async_cp16(kg + fo, &kb[sn][t]);
      async_cp16(vg + fo, &vb[sn][t]);
      wait_async<2>();     // 4 outstanding, in-order done => tile j landed
    } else {
      wait_async<0>();     // drain final tile
    }

    // Each lane consumes exactly the 16B it fetched: no barrier needed.
    float4 kk = kb[st][t];
    float4 vv = vb[st][t];
#pragma unroll
    for (int e = 0; e < 4; ++e) {
      float k   = (&kk.x)[e];
      float val = (&vv.x)[e];
      float d   = q - k;
      float a   = coef * d * d;            // <= 0
      float nm  = fmaxf(m, a);
      float rm  = fast_exp2(m - nm);
      float ra  = fast_exp2(a - nm);
      s = fmaf(s, rm, ra);
      v = fmaf(v, rm, ra * val);
      m = nm;
    }
  }

  // Wave32 tree reduction of (m, s, v).
#pragma unroll
  for (int off = 16; off > 0; off >>= 1) {
    float mo = __shfl_xor(m, off, 32);
    float so = __shfl_xor(s, off, 32);
    float vo = __shfl_xor(v, off, 32);
    nw_merge(m, s, v, mo, so, vo);
  }

  const int wave = t >> 5, lane = t & 31;
  if (lane == 0) { redm[wave] = m; reds[wave] = s; redv[wave] = v; }
  __syncthreads();

  if (t == 0) {
    float M = redm[0], S = reds[0], V = redv[0];
#pragma unroll
    for (int w = 1; w < TPB / 32; ++w) nw_merge(M, S, V, redm[w], reds[w], redv[w]);
    out[row] = V / S;
  }
}

extern "C" void kernel_launch(void* const* d_in, const int* in_sizes, int n_in,
                              void* d_out, int out_size, void* d_ws, size_t ws_size,
                              hipStream_t stream) {
  const float* key   = (const float*)d_in[0];   // [B, N]
  const float* value = (const float*)d_in[1];   // [B, N]
  const float* query = (const float*)d_in[2];   // [B]
  const float* sigma = (const float*)d_in[3];   // [1]
  float*       out   = (float*)d_out;           // [B]

  const int B = in_sizes[2];
  const int N = (int)((long long)in_sizes[0] / (long long)B);  // 16384

  nw_regression_kernel<<<dim3(B), dim3(TPB), 0, stream>>>(
      key, value, query, sigma, out, N);
}